// Wrapper_56495999812119
// MI455X (gfx1250) — compile-verified
//
#include <hip/hip_runtime.h>
#include <stdint.h>

typedef unsigned int u32;
typedef unsigned long long u64;

#define KTOP 1024
#define NCLS 80
#define ROWSZ 85

typedef __attribute__((ext_vector_type(2))) float v2f;
typedef __attribute__((ext_vector_type(8))) float v8f;

// monotone order-preserving map float -> u32 (larger float => larger key)
__device__ __forceinline__ u32 score_key(float f) {
  u32 u = __float_as_uint(f);
  return u ^ ((u & 0x80000000u) ? 0xFFFFFFFFu : 0x80000000u);
}

// ---------------- 1) per-row score key ----------------
__global__ void k_score(const float* __restrict__ pred, u32* __restrict__ keys, int n) {
  int i = blockIdx.x * blockDim.x + threadIdx.x;
  if (i >= n) return;
  const float* row = pred + (size_t)i * ROWSZ;
  float obj = row[4];
  float m = row[5];
#pragma unroll 8
  for (int c = 1; c < NCLS; ++c) m = fmaxf(m, row[5 + c]);
  float s = (obj * m >= 0.25f) ? obj : -1000000000.0f;
  keys[i] = score_key(s);
}

// ---------------- 2) init scratch (hist, sel, suppression mask) ----------------
// sel[0]=prefix, sel[1]=want, sel[2]=c_gt, sel[3]=c_eq
__global__ void k_init(u32* hist, u32* sel, u32* smask) {
  int t = blockIdx.x * blockDim.x + threadIdx.x;
  int stride = gridDim.x * blockDim.x;
  if (t < 256) hist[t] = 0u;
  if (t == 256) { sel[0] = 0u; sel[1] = (u32)KTOP; sel[2] = 0u; sel[3] = 0u; }
  for (int i = t; i < KTOP * 32; i += stride) smask[i] = 0u;
}

// ---------------- 3) radix-select histogram pass ----------------
__global__ void k_hist(const u32* __restrict__ keys, u32* __restrict__ hist,
                       const u32* __restrict__ sel, int pass, int n) {
  __shared__ u32 h[256];
  for (int b = threadIdx.x; b < 256; b += blockDim.x) h[b] = 0u;
  __syncthreads();
  int shift = 24 - 8 * pass;
  u32 prefix = sel[0];
  u32 mask = (pass == 0) ? 0u : (0xFFFFFFFFu << (shift + 8));
  int stride = gridDim.x * blockDim.x;
  for (int i = blockIdx.x * blockDim.x + threadIdx.x; i < n; i += stride) {
    u32 k = keys[i];
    if ((k & mask) == (prefix & mask)) atomicAdd(&h[(k >> shift) & 255u], 1u);
  }
  __syncthreads();
  for (int b = threadIdx.x; b < 256; b += blockDim.x)
    if (h[b]) atomicAdd(&hist[b], h[b]);
}

// ---------------- 4) radix-select pick pass (1 block) ----------------
__global__ void k_pick(u32* hist, u32* sel, int pass) {
  if (threadIdx.x == 0) {
    u32 want = sel[1];
    int shift = 24 - 8 * pass;
    u32 acc = 0;
    for (int b = 255; b >= 0; --b) {
      u32 c = hist[b];
      if (acc + c >= want) {
        sel[0] |= ((u32)b) << shift;
        sel[1] = want - acc;
        break;
      }
      acc += c;
    }
  }
  __syncthreads();
  for (int b = threadIdx.x; b < 256; b += blockDim.x) hist[b] = 0u;
}

// ---------------- 5) compaction of top-K candidates ----------------
__global__ void k_compact(const u32* __restrict__ keys, u32* sel,
                          u32* __restrict__ ckey, u32* __restrict__ cidx, int n) {
  u32 T = sel[0];
  u32 want = sel[1];
  u32 base = (u32)KTOP - want;
  int stride = gridDim.x * blockDim.x;
  for (int i = blockIdx.x * blockDim.x + threadIdx.x; i < n; i += stride) {
    u32 k = keys[i];
    if (k > T) {
      u32 p = atomicAdd(&sel[2], 1u);
      ckey[p] = k; cidx[p] = (u32)i;
    } else if (k == T) {
      u32 p = atomicAdd(&sel[3], 1u);
      if (p < want) { ckey[base + p] = k; cidx[base + p] = (u32)i; }
    }
  }
}

// ---------------- 6) bitonic sort (desc key, asc index) + gather ----------------
__global__ __launch_bounds__(1024) void k_sortgather(
    const u32* __restrict__ ckey, const u32* __restrict__ cidx,
    const float* __restrict__ pred,
    float* __restrict__ boxesk, float* __restrict__ areak,
    float* __restrict__ objk, float* __restrict__ confk,
    float* __restrict__ clsk, u32* __restrict__ validk) {
  __shared__ u64 s[KTOP];
  int t = threadIdx.x;
  s[t] = ((u64)ckey[t] << 32) | (u32)(~cidx[t]);
  __syncthreads();
  for (unsigned k = 2; k <= KTOP; k <<= 1) {
    for (unsigned j = k >> 1; j > 0; j >>= 1) {
      unsigned ixj = (unsigned)t ^ j;
      if (ixj > (unsigned)t) {
        u64 a = s[t], b = s[ixj];
        bool desc = (((unsigned)t & k) == 0u);
        bool sw = desc ? (a < b) : (a > b);
        if (sw) { s[t] = b; s[ixj] = a; }
      }
      __syncthreads();
    }
  }
  u64 v = s[t];
  u32 key = (u32)(v >> 32);
  u32 idx = ~((u32)v);
  const float* row = pred + (size_t)idx * ROWSZ;
  float cx = row[0], cy = row[1], w = row[2], h = row[3], obj = row[4];
  float best = row[5];
  int bestc = 0;
#pragma unroll 8
  for (int c = 1; c < NCLS; ++c) {
    float vv = row[5 + c];
    if (vv > best) { best = vv; bestc = c; }
  }
  float x1 = cx - w * 0.5f, y1 = cy - h * 0.5f;
  float x2 = cx + w * 0.5f, y2 = cy + h * 0.5f;
  boxesk[t * 4 + 0] = x1; boxesk[t * 4 + 1] = y1;
  boxesk[t * 4 + 2] = x2; boxesk[t * 4 + 3] = y2;
  areak[t] = (x2 - x1) * (y2 - y1);
  objk[t] = obj;
  confk[t] = best;
  clsk[t] = (float)bestc;
  validk[t] = (key >> 31) & 1u;   // score >= 0  <=>  > NEG/2
}

// ---------------- 7) pairwise IoU tiles: WMMA area-sum + ballot bit-pack ----------------
// One wave per 16x16 tile. WMMA computes area_i + area_j for the whole tile in a
// single V_WMMA_F32_16X16X4_F32; intersection is elementwise VALU; iou>thre bits
// are packed with wave32 ballots. Fully unrolled with static accumulator extracts
// so the WMMA is issued exactly once per tile.
#define IOU_ROW(R, BAL)                                                        \
  {                                                                            \
    const int rowl = rbase + (R);                                              \
    float lx = fmaxf(rb[rowl][0], cb[col][0]);                                 \
    float ly = fmaxf(rb[rowl][1], cb[col][1]);                                 \
    float hx = fminf(rb[rowl][2], cb[col][2]);                                 \
    float hy = fminf(rb[rowl][3], cb[col][3]);                                 \
    float iw = fmaxf(hx - lx, 0.0f);                                           \
    float ih = fmaxf(hy - ly, 0.0f);                                           \
    float inter = iw * ih;                                                     \
    float uni = c[R] - inter; /* area_i + area_j - inter (WMMA result) */      \
    float iou = inter / fmaxf(uni, 1e-9f);                                     \
    BAL = __builtin_amdgcn_ballot_w32(iou > 0.65f);                            \
  }

__global__ void k_iou(const float* __restrict__ boxesk, const float* __restrict__ areak,
                      u32* __restrict__ smask) {
  int i0 = blockIdx.y * 16;
  int j0 = blockIdx.x * 16;
  int lane = threadIdx.x;  // 32 = one wave
  __shared__ float rb[16][4];
  __shared__ float cb[16][4];
  if (lane < 16) {
#pragma unroll
    for (int cc = 0; cc < 4; ++cc) rb[lane][cc] = boxesk[(i0 + lane) * 4 + cc];
  } else {
    int l2 = lane - 16;
#pragma unroll
    for (int cc = 0; cc < 4; ++cc) cb[l2][cc] = boxesk[(j0 + l2) * 4 + cc];
  }
  __syncthreads();

  // A = [area_i, 1, 0, 0] (16x4), B = [1; area_j; 0; 0] (4x16)
  // => C[M][N] = area_row[M] + area_col[N]
  bool lo = (lane < 16);
  v2f a, b;
  a[0] = lo ? areak[i0 + lane] : 0.0f;           // K0 (lanes 0-15) / K2 (=0)
  a[1] = lo ? 1.0f : 0.0f;                       // K1 / K3
  b[0] = lo ? 1.0f : 0.0f;                       // B row K0 / K2
  b[1] = lo ? areak[j0 + (lane & 15)] : 0.0f;    // B row K1 / K3
  v8f c = {0.f, 0.f, 0.f, 0.f, 0.f, 0.f, 0.f, 0.f};
  c = __builtin_amdgcn_wmma_f32_16x16x4_f32(false, a, false, b, (short)0, c,
                                            false, false);

  const int col = lane & 15;
  const int rbase = lo ? 0 : 8;
  unsigned b0, b1, b2, b3, b4, b5, b6, b7;
  IOU_ROW(0, b0) IOU_ROW(1, b1) IOU_ROW(2, b2) IOU_ROW(3, b3)
  IOU_ROW(4, b4) IOU_ROW(5, b5) IOU_ROW(6, b6) IOU_ROW(7, b7)

  if (lane == 0) {
    const int wsh = j0 & 31;  // 0 or 16
    u32* base = smask + (size_t)i0 * 32 + (j0 >> 5);
    atomicOr(base + 0 * 32,  (b0 & 0xFFFFu) << wsh);
    atomicOr(base + 1 * 32,  (b1 & 0xFFFFu) << wsh);
    atomicOr(base + 2 * 32,  (b2 & 0xFFFFu) << wsh);
    atomicOr(base + 3 * 32,  (b3 & 0xFFFFu) << wsh);
    atomicOr(base + 4 * 32,  (b4 & 0xFFFFu) << wsh);
    atomicOr(base + 5 * 32,  (b5 & 0xFFFFu) << wsh);
    atomicOr(base + 6 * 32,  (b6 & 0xFFFFu) << wsh);
    atomicOr(base + 7 * 32,  (b7 & 0xFFFFu) << wsh);
    atomicOr(base + 8 * 32,  (b0 >> 16) << wsh);
    atomicOr(base + 9 * 32,  (b1 >> 16) << wsh);
    atomicOr(base + 10 * 32, (b2 >> 16) << wsh);
    atomicOr(base + 11 * 32, (b3 >> 16) << wsh);
    atomicOr(base + 12 * 32, (b4 >> 16) << wsh);
    atomicOr(base + 13 * 32, (b5 >> 16) << wsh);
    atomicOr(base + 14 * 32, (b6 >> 16) << wsh);
    atomicOr(base + 15 * 32, (b7 >> 16) << wsh);
  }
}

// ---------------- 8) serial greedy NMS on a single wave ----------------
__global__ void k_nms(const u32* __restrict__ smask, const u32* __restrict__ validk,
                      u32* __restrict__ keep) {
  int lane = threadIdx.x;  // 32 lanes; lane w owns kept-bit word w
  u32 kw = 0u;
  for (int i = 0; i < KTOP; ++i) {
    u32 rw = smask[i * 32 + lane];  // row i == column i (IoU symmetric)
    unsigned hit = __builtin_amdgcn_ballot_w32((rw & kw) != 0u);
    u32 v = validk[i];
    bool ki = (v != 0u) && (hit == 0u);
    if (ki && lane == (i >> 5)) kw |= (1u << (i & 31));
    if (lane == 0) keep[i] = ki ? 1u : 0u;
  }
}

// ---------------- 9) finalize outputs ----------------
__global__ void k_final(const float* __restrict__ boxesk, const float* __restrict__ objk,
                        const float* __restrict__ confk, const float* __restrict__ clsk,
                        const u32* __restrict__ keep, float* __restrict__ out) {
  int t = blockIdx.x * blockDim.x + threadIdx.x;
  if (t >= KTOP) return;
  float kf = keep[t] ? 1.0f : 0.0f;
  out[t * 7 + 0] = boxesk[t * 4 + 0] * kf;
  out[t * 7 + 1] = boxesk[t * 4 + 1] * kf;
  out[t * 7 + 2] = boxesk[t * 4 + 2] * kf;
  out[t * 7 + 3] = boxesk[t * 4 + 3] * kf;
  out[t * 7 + 4] = objk[t] * kf;
  out[t * 7 + 5] = confk[t] * kf;
  out[t * 7 + 6] = clsk[t] * kf;
  out[KTOP * 7 + t] = kf;  // keep as float 0/1
}

extern "C" void kernel_launch(void* const* d_in, const int* in_sizes, int n_in,
                              void* d_out, int out_size, void* d_ws, size_t ws_size,
                              hipStream_t stream) {
  const float* pred = (const float*)d_in[0];
  int n = in_sizes[0] / ROWSZ;
  float* out = (float*)d_out;

  // workspace layout
  char* ws = (char*)d_ws;
  size_t off = 0;
  u32* keys = (u32*)(ws + off); off += (size_t)n * 4;
  u32* hist = (u32*)(ws + off); off += 256 * 4;
  u32* sel  = (u32*)(ws + off); off += 64;
  u32* ckey = (u32*)(ws + off); off += KTOP * 4;
  u32* cidx = (u32*)(ws + off); off += KTOP * 4;
  float* boxesk = (float*)(ws + off); off += KTOP * 4 * 4;
  float* areak  = (float*)(ws + off); off += KTOP * 4;
  float* objk   = (float*)(ws + off); off += KTOP * 4;
  float* confk  = (float*)(ws + off); off += KTOP * 4;
  float* clsk   = (float*)(ws + off); off += KTOP * 4;
  u32* validk   = (u32*)(ws + off); off += KTOP * 4;
  u32* smask    = (u32*)(ws + off); off += (size_t)KTOP * 32 * 4;
  u32* keep     = (u32*)(ws + off); off += KTOP * 4;

  k_score<<<(n + 255) / 256, 256, 0, stream>>>(pred, keys, n);
  k_init<<<128, 256, 0, stream>>>(hist, sel, smask);
  for (int pass = 0; pass < 4; ++pass) {
    k_hist<<<512, 256, 0, stream>>>(keys, hist, sel, pass, n);
    k_pick<<<1, 256, 0, stream>>>(hist, sel, pass);
  }
  k_compact<<<512, 256, 0, stream>>>(keys, sel, ckey, cidx, n);
  k_sortgather<<<1, 1024, 0, stream>>>(ckey, cidx, pred, boxesk, areak, objk,
                                       confk, clsk, validk);
  k_iou<<<dim3(64, 64), 32, 0, stream>>>(boxesk, areak, smask);
  k_nms<<<1, 32, 0, stream>>>(smask, validk, keep);
  k_final<<<(KTOP + 255) / 256, 256, 0, stream>>>(boxesk, objk, confk, clsk, keep, out);
}